// Attention_45870250721399
// MI455X (gfx1250) — compile-verified
//
#include <hip/hip_runtime.h>
#include <hip/hip_bf16.h>

// ---------------------------------------------------------------------------
// Bahdanau (MLP) attention for MI455X / gfx1250, fp32 end-to-end.
// GEMMs: V_WMMA_F32_16X16X4_F32, one wave per 32x64 tile, double-buffered
// fragment loads so VMEM overlaps the matrix pipe (no wait-0 per WMMA).
// Score stage: hardware v_tanh_f32 + global_prefetch_b8 row prefetch.
// ---------------------------------------------------------------------------

typedef __attribute__((ext_vector_type(2))) float v2f;
typedef __attribute__((ext_vector_type(8))) float v8f;

static constexpr int Bn = 4;
static constexpr int Tn = 128;
static constexpr int Sn = 512;
static constexpr int Hn = 512;

// ---------------- transcendental helpers (CDNA5 hardware paths) ------------
__device__ __forceinline__ float fast_tanh(float x) {
#if __has_builtin(__builtin_amdgcn_tanhf)
  return __builtin_amdgcn_tanhf(x);           // v_tanh_f32
#elif __has_builtin(__builtin_amdgcn_tanh_f32)
  return __builtin_amdgcn_tanh_f32(x);
#else
  return tanhf(x);
#endif
}

__device__ __forceinline__ float fast_exp2(float x) {
#if __has_builtin(__builtin_amdgcn_exp2f)
  return __builtin_amdgcn_exp2f(x);           // v_exp_f32
#else
  return exp2f(x);
#endif
}

#define WMMA_F32(a, b, c) \
  __builtin_amdgcn_wmma_f32_16x16x4_f32(false, (a), false, (b), (short)0, (c), false, false)

// ---------------- WMMA f32 16x16x4 fragment helpers ------------------------
// A-frag (16x4 f32): lane L -> M = L%16, K = 2*(L/16) + {0,1}.
// B-frag (4x16 f32): lane L -> N = L%16, K = 2*(L/16) + {0,1}.
// C/D (16x16 f32):  VGPR r, lanes 0-15 -> (M=r, N=lane); lanes 16-31 -> (M=r+8).
__device__ __forceinline__ void load_frag_a(const float* arow0, const float* arow1,
                                            int k, int koff, v2f& a0, v2f& a1) {
  a0.x = arow0[k + koff];
  a0.y = arow0[k + koff + 1];
  a1.x = arow1[k + koff];
  a1.y = arow1[k + koff + 1];
}

__device__ __forceinline__ void load_frag_b(const float* Bp, int sK, int sN,
                                            int lm, int k, int koff, v2f b[4]) {
#pragma unroll
  for (int j = 0; j < 4; ++j) {
    const float* bp = Bp + (k + koff) * sK + (j * 16 + lm) * sN;
    b[j].x = bp[0];
    b[j].y = bp[sK];
  }
}

// One wave accumulates a 32(M) x 64(N) tile with double-buffered fragments:
// 8 WMMAs per K-step of 4, next-stage loads issued before current WMMAs.
// K must be a multiple of 8 (here K is 512).
__device__ __forceinline__ void wmma_accum_tile32(
    const float* __restrict__ A, int lda,          // A tile base (row0, k=0)
    const float* __restrict__ Bp, int sK, int sN,  // B(k,n) = Bp[k*sK + n*sN]
    int K, int lane, v8f acc[2][4])
{
  const int lm   = lane & 15;
  const int koff = (lane >> 4) << 1;               // 0 or 2
  const float* arow0 = A + lm * lda;
  const float* arow1 = A + (16 + lm) * lda;

  v2f aA0, aA1, bA[4];                             // stage buffer A
  v2f aB0, aB1, bB[4];                             // stage buffer B
  load_frag_a(arow0, arow1, 0, koff, aA0, aA1);
  load_frag_b(Bp, sK, sN, lm, 0, koff, bA);

  int k0 = 0;
  for (; k0 < K - 8; k0 += 8) {
    // prefetch stage k0+4 while computing k0
    load_frag_a(arow0, arow1, k0 + 4, koff, aB0, aB1);
    load_frag_b(Bp, sK, sN, lm, k0 + 4, koff, bB);
#pragma unroll
    for (int j = 0; j < 4; ++j) {
      acc[0][j] = WMMA_F32(aA0, bA[j], acc[0][j]);
      acc[1][j] = WMMA_F32(aA1, bA[j], acc[1][j]);
    }
    // prefetch stage k0+8 while computing k0+4
    load_frag_a(arow0, arow1, k0 + 8, koff, aA0, aA1);
    load_frag_b(Bp, sK, sN, lm, k0 + 8, koff, bA);
#pragma unroll
    for (int j = 0; j < 4; ++j) {
      acc[0][j] = WMMA_F32(aB0, bB[j], acc[0][j]);
      acc[1][j] = WMMA_F32(aB1, bB[j], acc[1][j]);
    }
  }
  // epilogue: k0 == K-8; current buffer A holds stage K-8
  load_frag_a(arow0, arow1, k0 + 4, koff, aB0, aB1);
  load_frag_b(Bp, sK, sN, lm, k0 + 4, koff, bB);
#pragma unroll
  for (int j = 0; j < 4; ++j) {
    acc[0][j] = WMMA_F32(aA0, bA[j], acc[0][j]);
    acc[1][j] = WMMA_F32(aA1, bA[j], acc[1][j]);
  }
#pragma unroll
  for (int j = 0; j < 4; ++j) {
    acc[0][j] = WMMA_F32(aB0, bB[j], acc[0][j]);
    acc[1][j] = WMMA_F32(aB1, bB[j], acc[1][j]);
  }
}

__device__ __forceinline__ void wmma_store_tile16(
    float* __restrict__ C, int ldo, int lane,
    const v8f acc[4], const float* __restrict__ bias) // bias at tile col0 (or null)
{
  const int lm    = lane & 15;
  const int rbase = (lane >> 4) * 8;
#pragma unroll
  for (int j = 0; j < 4; ++j) {
    const int col = j * 16 + lm;
    const float bv = bias ? bias[col] : 0.0f;
#pragma unroll
    for (int r = 0; r < 8; ++r) {
      C[(rbase + r) * ldo + col] = acc[j][r] + bv;
    }
  }
}

__device__ __forceinline__ void wmma_store_tile32(
    float* __restrict__ C, int ldo, int lane,
    const v8f acc[2][4], const float* __restrict__ bias)
{
  wmma_store_tile16(C, ldo, lane, acc[0], bias);
  wmma_store_tile16(C + 16 * ldo, ldo, lane, acc[1], bias);
}

#define ZERO_ACC(acc)                                     \
  v8f acc[2][4];                                          \
  {                                                       \
    v8f z_ = {0.f,0.f,0.f,0.f,0.f,0.f,0.f,0.f};           \
    _Pragma("unroll") for (int i_ = 0; i_ < 2; ++i_)      \
      _Pragma("unroll") for (int j_ = 0; j_ < 4; ++j_)    \
        acc[i_][j_] = z_;                                 \
  }

// ---------------- Kernel 1/2: C[M,N] = A[M,K] @ W[N,K]^T (+ bias) ----------
__global__ __launch_bounds__(256) void gemm_wt_kernel(
    const float* __restrict__ A, const float* __restrict__ W,
    const float* __restrict__ bias, float* __restrict__ C,
    int M, int N, int K)
{
  const int wave = (blockIdx.x * blockDim.x + threadIdx.x) >> 5;
  const int lane = threadIdx.x & 31;
  const int ntn  = N >> 6;
  const int tm   = wave / ntn;                    // 32-row tiles
  const int tn   = wave % ntn;
  if (tm * 32 >= M) return;                       // wave-uniform, EXEC stays full

  ZERO_ACC(acc);
  // B(k,n) = W[(col0+n)*K + k]  -> sK=1, sN=K
  wmma_accum_tile32(A + tm * 32 * K, K, W + (long)(tn * 64) * K, 1, K, K, lane, acc);
  wmma_store_tile32(C + tm * 32 * N + tn * 64, N, lane, acc,
                    bias ? bias + tn * 64 : nullptr);
}

// ---------------- Kernel 3: scores + softmax (fused) -----------------------
// One block per (b,t). Each wave handles score rows s = wave, wave+8, ...
// align is written directly in the transposed [T,B,S] output layout.
__global__ __launch_bounds__(256) void score_softmax_kernel(
    const float* __restrict__ wq,     // [B*T, H]
    const float* __restrict__ uh,     // [B*S, H]
    const float* __restrict__ v,      // [H]
    float* __restrict__ align_t)      // [T, B, S]
{
  __shared__ float s_wq[Hn];
  __shared__ float s_v[Hn];
  __shared__ float s_sc[Sn];
  __shared__ float red[8];

  const int bt   = blockIdx.x;
  const int b    = bt / Tn;
  const int t    = bt % Tn;
  const int tid  = threadIdx.x;
  const int wave = tid >> 5;
  const int lane = tid & 31;

  for (int h = tid; h < Hn; h += 256) {
    s_wq[h] = wq[bt * Hn + h];
    s_v[h]  = v[h];
  }
  __syncthreads();

  // scores: one wave per s row -> coalesced uh reads, 16 tanh+fma per lane.
  // Prefetch the next s row (2KB spread over 32 lanes) while computing.
  for (int s = wave; s < Sn; s += 8) {
    const float* up = uh + (b * Sn + s) * Hn;
    const int snext = (s + 8 < Sn) ? (s + 8) : s;
    const float* upn = uh + (b * Sn + snext) * Hn;
    __builtin_prefetch(upn + lane * 16, 0, 1);     // global_prefetch_b8
    float acc = 0.f;
    for (int h = lane; h < Hn; h += 32) {
      acc = fmaf(s_v[h], fast_tanh(s_wq[h] + up[h]), acc);
    }
#pragma unroll
    for (int off = 16; off > 0; off >>= 1)
      acc += __shfl_down(acc, off, 32);
    if (lane == 0) s_sc[s] = acc;
  }
  __syncthreads();

  // row max
  float m = -1e30f;
  for (int s = tid; s < Sn; s += 256) m = fmaxf(m, s_sc[s]);
#pragma unroll
  for (int off = 16; off > 0; off >>= 1)
    m = fmaxf(m, __shfl_down(m, off, 32));
  if (lane == 0) red[wave] = m;
  __syncthreads();
  if (tid == 0) {
    float mm = red[0];
#pragma unroll
    for (int i = 1; i < 8; ++i) mm = fmaxf(mm, red[i]);
    red[0] = mm;
  }
  __syncthreads();
  const float rowmax = red[0];
  __syncthreads();

  // exp (exp2 with log2e scaling -> v_exp_f32) and sum
  float psum = 0.f;
  for (int s = tid; s < Sn; s += 256) {
    float e = fast_exp2((s_sc[s] - rowmax) * 1.4426950408889634f);
    s_sc[s] = e;
    psum += e;
  }
#pragma unroll
  for (int off = 16; off > 0; off >>= 1)
    psum += __shfl_down(psum, off, 32);
  if (lane == 0) red[wave] = psum;
  __syncthreads();
  if (tid == 0) {
    float ss = 0.f;
#pragma unroll
    for (int i = 0; i < 8; ++i) ss += red[i];
    red[0] = ss;
  }
  __syncthreads();
  const float inv = 1.0f / red[0];

  float* orow = align_t + (t * Bn + b) * Sn;       // coalesced over s
  for (int s = tid; s < Sn; s += 256) orow[s] = s_sc[s] * inv;
}

// ---------------- Kernel 4: c[b,t,h] = Σ_s align[b,t,s] * enc[b,s,h] -------
__global__ __launch_bounds__(256) void gemm_ctx_kernel(
    const float* __restrict__ align_t, // [T, B, S]
    const float* __restrict__ enc,     // [B, S, H]
    float* __restrict__ Cout)          // [B, T, H]
{
  const int wave = (blockIdx.x * blockDim.x + threadIdx.x) >> 5;
  const int lane = threadIdx.x & 31;
  const int tiles_per_b = (Tn / 32) * (Hn / 64);   // 4 * 8 = 32
  const int b  = wave / tiles_per_b;
  const int r  = wave % tiles_per_b;
  const int tm = r / (Hn / 64);
  const int tn = r % (Hn / 64);

  ZERO_ACC(acc);
  // A(t,s) = align_t[t*B*S + b*S + s]  -> lda = B*S
  // B(s,h) = enc[b*S*H + s*H + h]      -> sK = H, sN = 1
  wmma_accum_tile32(align_t + tm * 32 * (Bn * Sn) + b * Sn, Bn * Sn,
                    enc + (long)b * Sn * Hn + tn * 64, Hn, 1,
                    Sn, lane, acc);
  wmma_store_tile32(Cout + ((long)b * Tn + tm * 32) * Hn + tn * 64, Hn, lane, acc,
                    nullptr);
}

// ---------------- Kernel 5: attn = [c,dec] @ Wo^T + bo  (two K passes) -----
__global__ __launch_bounds__(256) void gemm_out_kernel(
    const float* __restrict__ Cctx,  // [B*T, H]
    const float* __restrict__ dec,   // [B*T, H]
    const float* __restrict__ Wo,    // [H, 2H] row-major
    const float* __restrict__ bo,    // [H]
    float* __restrict__ out)         // [B*T, H]
{
  const int wave = (blockIdx.x * blockDim.x + threadIdx.x) >> 5;
  const int lane = threadIdx.x & 31;
  const int ntn  = Hn / 64;
  const int tm   = wave / ntn;
  const int tn   = wave % ntn;

  ZERO_ACC(acc);
  // concat handled as two K-passes over Wo's left/right halves:
  // B(k,n) = Wo[(col0+n)*2H + k]        (c part)
  wmma_accum_tile32(Cctx + tm * 32 * Hn, Hn,
                    Wo + (long)(tn * 64) * (2 * Hn), 1, 2 * Hn, Hn, lane, acc);
  // B(k,n) = Wo[(col0+n)*2H + H + k]    (dec part)
  wmma_accum_tile32(dec + tm * 32 * Hn, Hn,
                    Wo + (long)(tn * 64) * (2 * Hn) + Hn, 1, 2 * Hn, Hn, lane, acc);
  wmma_store_tile32(out + tm * 32 * Hn + tn * 64, Hn, lane, acc, bo + tn * 64);
}

// ---------------------------------------------------------------------------
extern "C" void kernel_launch(void* const* d_in, const int* in_sizes, int n_in,
                              void* d_out, int out_size, void* d_ws, size_t ws_size,
                              hipStream_t stream)
{
  const float* dec = (const float*)d_in[0];  // [B,T,H]
  const float* enc = (const float*)d_in[1];  // [B,S,H]
  const float* Wq  = (const float*)d_in[2];  // [H,H]
  const float* bq  = (const float*)d_in[3];  // [H]
  const float* Wc  = (const float*)d_in[4];  // [H,H]
  const float* v   = (const float*)d_in[5];  // [H]
  const float* Wo  = (const float*)d_in[6];  // [H,2H]
  const float* bo  = (const float*)d_in[7];  // [H]

  float* attn    = (float*)d_out;                 // [B,T,H]
  float* align_t = (float*)d_out + Bn * Tn * Hn;  // [T,B,S]

  float* wq   = (float*)d_ws;                     // [B*T, H]   1 MB
  float* uh   = wq + Bn * Tn * Hn;                // [B*S, H]   4 MB
  float* cctx = uh + Bn * Sn * Hn;                // [B*T, H]   1 MB

  // 1) wq = dec @ Wq^T + bq : 16 m-tiles(32) * 8 n-tiles(64) = 128 waves
  gemm_wt_kernel<<<16, 256, 0, stream>>>(dec, Wq, bq, wq, Bn * Tn, Hn, Hn);
  // 2) uh = enc @ Wc^T      : 64 * 8 = 512 waves
  gemm_wt_kernel<<<64, 256, 0, stream>>>(enc, Wc, nullptr, uh, Bn * Sn, Hn, Hn);
  // 3) scores + softmax, writes align in transposed [T,B,S] layout
  score_softmax_kernel<<<Bn * Tn, 256, 0, stream>>>(wq, uh, v, align_t);
  // 4) c = align @ enc      : 4 batches * 32 tiles = 128 waves
  gemm_ctx_kernel<<<16, 256, 0, stream>>>(align_t, enc, cctx);
  // 5) attn = [c,dec] @ Wo^T + bo : 128 waves
  gemm_out_kernel<<<16, 256, 0, stream>>>(cctx, dec, Wo, bo, attn);
}